// diff_lpc2rc_52664888984120
// MI455X (gfx1250) — compile-verified
//
#include <hip/hip_runtime.h>
#include <cstdint>

#ifndef __has_builtin
#define __has_builtin(x) 0
#endif

#define AS1 __attribute__((address_space(1)))
#define AS3 __attribute__((address_space(3)))

typedef int v4i __attribute__((ext_vector_type(4)));

constexpr int TPB          = 256;   // threads per block = rows per tile
constexpr int ROWF         = 16;    // floats per row
constexpr int ROW_STRIDE_B = 80;    // 64B of data + 16B pad (keeps b128 writes 16B-aligned)
constexpr int BUF_B        = TPB * ROW_STRIDE_B;   // 20480 B per buffer

// ---------------- CDNA5 async global->LDS copy (ASYNCcnt-tracked) ----------------
__device__ __forceinline__ void async_copy_b128(const void* g, void* l) {
#if __has_builtin(__builtin_amdgcn_global_load_async_to_lds_b128)
  __builtin_amdgcn_global_load_async_to_lds_b128((AS1 v4i*)g, (AS3 v4i*)l, 0, 0);
#else
  uint32_t loff = (uint32_t)(uintptr_t)(AS3 void*)l;
  asm volatile("global_load_async_to_lds_b128 %0, %1, off"
               :: "v"(loff), "v"(g) : "memory");
#endif
}

template <int CNT>
__device__ __forceinline__ void wait_asynccnt() {
#if __has_builtin(__builtin_amdgcn_s_wait_asynccnt)
  __builtin_amdgcn_s_wait_asynccnt(CNT);
#else
  asm volatile("s_wait_asynccnt %0" :: "i"(CNT) : "memory");
#endif
}

// ---------------- per-row lpc2rc recursion, fully unrolled in VGPRs ----------------
__device__ __forceinline__ void lpc2rc_row(float a[ROWF]) {
#pragma unroll
  for (int m = ROWF - 1; m >= 1; --m) {
    const float ki  = a[m];
    const float inv = 1.0f / (1.0f - ki * ki);
#pragma unroll
    for (int t = 0; t < m / 2; ++t) {
      const float lo = a[t];
      const float hi = a[m - 1 - t];
      a[t]         = (lo - ki * hi) * inv;
      a[m - 1 - t] = (hi - ki * lo) * inv;
    }
    if (m & 1) {
      const int mid = m >> 1;
      a[mid] = (a[mid] - ki * a[mid]) * inv;
    }
  }
}

// ---------------- main pipelined kernel: async LDS double buffering ----------------
__global__ __launch_bounds__(TPB) void lpc2rc_pipe(const float* __restrict__ x,
                                                   float* __restrict__ out,
                                                   int ntiles) {
  __shared__ alignas(16) unsigned char smem[2 * BUF_B];
  const int tid = threadIdx.x;
  const int nb  = gridDim.x;

  // Issue the 4 async b128 chunks that cover one 256-row tile into buffer `buf`.
  // Chunk ci = i*TPB + tid -> row ci>>2, quarter ci&3. Consecutive lanes hit
  // consecutive 16B chunks of global memory: fully coalesced 512B per wave.
  auto issue_tile = [&](int tile, int buf) {
    const float* base = x + (size_t)tile * TPB * ROWF;
#pragma unroll
    for (int i = 0; i < 4; ++i) {
      const int ci  = i * TPB + tid;
      const int row = ci >> 2;
      const int q   = ci & 3;
      const float* g   = base + (size_t)row * ROWF + q * 4;
      unsigned char* l = smem + buf * BUF_B + row * ROW_STRIDE_B + q * 16;
      async_copy_b128(g, l);
    }
  };

  int t = blockIdx.x;
  if (t < ntiles) issue_tile(t, 0);

  int parity = 0;
  for (; t < ntiles; t += nb) {
    const int tn = t + nb;
    if (tn < ntiles) {           // prefetch next tile into the other buffer
      issue_tile(tn, parity ^ 1);
      wait_asynccnt<4>();        // in-order completion: oldest 4 (tile t) are done
    } else {
      wait_asynccnt<0>();
    }
    __syncthreads();             // tile t fully visible in LDS to all waves

    // Pull this thread's row out of LDS (4 x ds_load_b128, 16B-aligned).
    float a[ROWF];
    const float4* lp =
        reinterpret_cast<const float4*>(smem + parity * BUF_B + tid * ROW_STRIDE_B);
#pragma unroll
    for (int i = 0; i < 4; ++i) {
      const float4 v = lp[i];
      a[4 * i + 0] = v.x; a[4 * i + 1] = v.y; a[4 * i + 2] = v.z; a[4 * i + 3] = v.w;
    }

    lpc2rc_row(a);

    float4* op = reinterpret_cast<float4*>(out + ((size_t)t * TPB + tid) * ROWF);
#pragma unroll
    for (int i = 0; i < 4; ++i) {
      float4 v;
      v.x = a[4 * i + 0]; v.y = a[4 * i + 1]; v.z = a[4 * i + 2]; v.w = a[4 * i + 3];
      op[i] = v;
    }

    __syncthreads();             // everyone done reading buf `parity` before reuse
    parity ^= 1;
  }
}

// ---------------- tail kernel for rows beyond the last full tile ----------------
__global__ __launch_bounds__(TPB) void lpc2rc_tail(const float* __restrict__ x,
                                                   float* __restrict__ out,
                                                   long long startRow,
                                                   long long totalRows) {
  const long long r = startRow + (long long)blockIdx.x * blockDim.x + threadIdx.x;
  if (r >= totalRows) return;
  float a[ROWF];
  const float4* ip = reinterpret_cast<const float4*>(x + r * ROWF);
#pragma unroll
  for (int i = 0; i < 4; ++i) {
    const float4 v = ip[i];
    a[4 * i + 0] = v.x; a[4 * i + 1] = v.y; a[4 * i + 2] = v.z; a[4 * i + 3] = v.w;
  }
  lpc2rc_row(a);
  float4* op = reinterpret_cast<float4*>(out + r * ROWF);
#pragma unroll
  for (int i = 0; i < 4; ++i) {
    float4 v;
    v.x = a[4 * i + 0]; v.y = a[4 * i + 1]; v.z = a[4 * i + 2]; v.w = a[4 * i + 3];
    op[i] = v;
  }
}

extern "C" void kernel_launch(void* const* d_in, const int* in_sizes, int n_in,
                              void* d_out, int out_size, void* d_ws, size_t ws_size,
                              hipStream_t stream) {
  (void)n_in; (void)out_size; (void)d_ws; (void)ws_size;
  const float* x = (const float*)d_in[0];
  float* out     = (float*)d_out;

  const long long totalRows = (long long)in_sizes[0] / ROWF;  // 64*32768 = 2,097,152
  const int ntiles = (int)(totalRows / TPB);                  // 8192 full tiles

  if (ntiles > 0) {
    const int grid = ntiles < 2048 ? ntiles : 2048;           // ~4 tiles/block: double buffering pays off
    lpc2rc_pipe<<<grid, TPB, 0, stream>>>(x, out, ntiles);
  }
  const long long rem = totalRows - (long long)ntiles * TPB;
  if (rem > 0) {
    const int grid = (int)((rem + TPB - 1) / TPB);
    lpc2rc_tail<<<grid, TPB, 0, stream>>>(x, out, (long long)ntiles * TPB, totalRows);
  }
}